// Voxels_5669356833119
// MI455X (gfx1250) — compile-verified
//
#include <hip/hip_runtime.h>

typedef __attribute__((ext_vector_type(2))) float v2f;
typedef __attribute__((ext_vector_type(4))) float v4f;
typedef __attribute__((ext_vector_type(8))) float v8f;

#define VOX_SIDE   256
#define CH_STRIDE  (VOX_SIDE * VOX_SIDE * VOX_SIDE)   // 16,777,216 floats / channel

// One wave (32 lanes) handles 16 points. Lanes 0-15 compute channels 0,1 of
// point (base+lane); lanes 16-31 compute channels 2,3 of point (base+lane-16).
// This matches the 4x16 B-matrix VGPR layout of V_WMMA_F32_16X16X4_F32 exactly,
// so a single WMMA with an identity A performs the channel->point transpose
// plus the bias add (C = bias rows), giving each of lanes 0-15 a ready float4.
__global__ __launch_bounds__(256) void voxel_trilerp_wmma(
    const float* __restrict__ pos,   // (N,3)
    const float* __restrict__ vox,   // (4,256,256,256)
    const float* __restrict__ bias,  // (1,4)
    float*       __restrict__ out,   // (N,4)
    int npts)
{
    const int tid   = threadIdx.x;
    const int lane  = tid & 31;
    const int wv    = tid >> 5;
    const int m     = lane & 15;
    const int pbase = (blockIdx.x * (256 / 32) + wv) * 16;
    const int p     = pbase + m;
    const int pc    = min(p, npts - 1);     // clamp instead of mask: EXEC stays all-ones
    const int chPr  = lane >> 4;            // 0 -> channels 0,1 ; 1 -> channels 2,3

    // Positions: streaming read-once -> non-temporal, keep L2 for the voxel grid.
    const float* pp = pos + (size_t)3 * (size_t)pc;
    const float gx = __builtin_nontemporal_load(pp + 0);
    const float gy = __builtin_nontemporal_load(pp + 1);
    const float gz = __builtin_nontemporal_load(pp + 2);

    // src index, align_corners=False, border clamp (SCALE == 1.0)
    float ix = ((gx + 1.0f) * 256.0f - 1.0f) * 0.5f;
    float iy = ((gy + 1.0f) * 256.0f - 1.0f) * 0.5f;
    float iz = ((gz + 1.0f) * 256.0f - 1.0f) * 0.5f;
    ix = fminf(fmaxf(ix, 0.0f), 255.0f);
    iy = fminf(fmaxf(iy, 0.0f), 255.0f);
    iz = fminf(fmaxf(iz, 0.0f), 255.0f);

    const float fx = floorf(ix), fy = floorf(iy), fz = floorf(iz);
    const float wx = ix - fx, wy = iy - fy, wz = iz - fz;
    const float ux = 1.0f - wx, uy = 1.0f - wy, uz = 1.0f - wz;

    const int x0 = (int)fx, y0 = (int)fy, z0 = (int)fz;
    const int x1 = min(x0 + 1, 255);
    const int y1 = min(y0 + 1, 255);
    const int z1 = min(z0 + 1, 255);

    const int zy00 = (z0 << 16) + (y0 << 8);
    const int zy01 = (z0 << 16) + (y1 << 8);
    const int zy10 = (z1 << 16) + (y0 << 8);
    const int zy11 = (z1 << 16) + (y1 << 8);
    const int o0 = zy00 + x0, o1 = zy00 + x1;
    const int o2 = zy01 + x0, o3 = zy01 + x1;
    const int o4 = zy10 + x0, o5 = zy10 + x1;
    const int o6 = zy11 + x0, o7 = zy11 + x1;

    const float uyux = uy * ux, uywx = uy * wx, wyux = wy * ux, wywx = wy * wx;
    const float w0 = uz * uyux, w1 = uz * uywx, w2 = uz * wyux, w3 = uz * wywx;
    const float w4 = wz * uyux, w5 = wz * uywx, w6 = wz * wyux, w7 = wz * wywx;

    // 16 independent gathers per lane (2 channels x 8 corners): issue all loads
    // first so LOADcnt covers the full batch and HBM latency is overlapped.
    const float* c0 = vox + (size_t)(chPr * 2) * (size_t)CH_STRIDE;
    const float* c1 = c0 + CH_STRIDE;
    const float v00 = c0[o0], v01 = c0[o1], v02 = c0[o2], v03 = c0[o3];
    const float v04 = c0[o4], v05 = c0[o5], v06 = c0[o6], v07 = c0[o7];
    const float v10 = c1[o0], v11 = c1[o1], v12 = c1[o2], v13 = c1[o3];
    const float v14 = c1[o4], v15 = c1[o5], v16 = c1[o6], v17 = c1[o7];

    const float s0 = fmaf(w0, v00, fmaf(w1, v01, fmaf(w2, v02, fmaf(w3, v03,
                     fmaf(w4, v04, fmaf(w5, v05, fmaf(w6, v06, w7 * v07)))))));
    const float s1 = fmaf(w0, v10, fmaf(w1, v11, fmaf(w2, v12, fmaf(w3, v13,
                     fmaf(w4, v14, fmaf(w5, v15, fmaf(w6, v16, w7 * v17)))))));

    // A = identity (16x4, MxK): ISA layout -> VGPR0 holds K=0 (lanes 0-15) and
    // K=2 (lanes 16-31); VGPR1 holds K=1 / K=3. Identity => 1.0 at lanes 0,18 / 1,19.
    v2f A;
    A.x = (lane == 0 || lane == 18) ? 1.0f : 0.0f;
    A.y = (lane == 1 || lane == 19) ? 1.0f : 0.0f;

    // B = sampled values (4x16, KxN): VGPR0 = row K=0 (lanes 0-15) / K=2 (16-31),
    // VGPR1 = row K=1 / K=3 — exactly our per-lane channel pair.
    v2f B;
    B.x = s0;
    B.y = s1;

    // C[m,n] = bias[m] for the valid rows (m<4, lanes 0-15); rest zero.
    v8f C = {};
    if (lane < 16) {
        C[0] = bias[0]; C[1] = bias[1]; C[2] = bias[2]; C[3] = bias[3];
    }

    // D = I x B + C : matrix pipe does the channel->point transpose + bias add.
    v8f D = __builtin_amdgcn_wmma_f32_16x16x4_f32(
        false, A, false, B, (short)0, C, false, false);

    // D VGPR r, lanes 0-15: element (M=r, N=lane) = channel r of point pbase+lane.
    if (lane < 16 && p < npts) {
        v4f r;
        r.x = D[0]; r.y = D[1]; r.z = D[2]; r.w = D[3];
        __builtin_nontemporal_store(r, (v4f*)out + p);  // streaming NT store, spare L2
    }
}

extern "C" void kernel_launch(void* const* d_in, const int* in_sizes, int n_in,
                              void* d_out, int out_size, void* d_ws, size_t ws_size,
                              hipStream_t stream)
{
    const float* pos  = (const float*)d_in[0];
    const float* vox  = (const float*)d_in[1];
    const float* bias = (const float*)d_in[2];
    float* out = (float*)d_out;

    const int npts = in_sizes[0] / 3;          // 2,097,152
    const int points_per_block = (256 / 32) * 16;  // 8 waves x 16 points = 128
    const int blocks = (npts + points_per_block - 1) / points_per_block;

    voxel_trilerp_wmma<<<blocks, 256, 0, stream>>>(pos, vox, bias, out, npts);
}